// MRTGAutoencoder_83107617178203
// MI455X (gfx1250) — compile-verified
//
#include <hip/hip_runtime.h>
#include <hip/hip_bf16.h>
#include <math.h>

// ---------------------------------------------------------------------------
// MRTG autoencoder forward loss on MI455X (gfx1250, wave32).
// Heavy GEMMs use V_WMMA_F32_16X16X4_F32 (exact fp32 matrix pipeline).
// 32x64 per-wave register tiles; split-K=4 (deterministic) for the skinny
// alpha^T @ x GEMMs; global_prefetch_b8 on the 64MB streamed operand.
// ---------------------------------------------------------------------------

constexpr int NPTS = 4096;   // n
constexpr int PDIM = 128;    // p
constexpr int LDIM = 32;     // L
constexpr int KSPLIT = 4;    // split-K factor for alpha^T @ x

typedef float v2f __attribute__((ext_vector_type(2)));
typedef float v8f __attribute__((ext_vector_type(8)));

__device__ __forceinline__ v8f wmma4(v2f a, v2f b, v8f c) {
  // D = A(16x4,f32) * B(4x16,f32) + C(16x16,f32)
  return __builtin_amdgcn_wmma_f32_16x16x4_f32(
      /*neg_a=*/false, a, /*neg_b=*/false, b,
      /*c_mod=*/(short)0, c, /*reuse_a=*/false, /*reuse_b=*/false);
}

__device__ __forceinline__ v8f v8zero() {
  v8f z;
#pragma unroll
  for (int i = 0; i < 8; ++i) z[i] = 0.0f;
  return z;
}

__device__ __forceinline__ float w1f(float d, float r) {
  float t = d / r;
  float base = (d < r) ? (1.0f - t * t) : 0.0f;
  return base * base * base;
}

__device__ __forceinline__ float w2f(float d, float r) {
  if (d < 0.5f * r) return 1.0f;
  float t = (2.0f * d - r) / r;
  float base = (d < r) ? (1.0f - t * t) : 0.0f;
  return base * base * base;
}

__device__ __forceinline__ float gelu_exact(float v) {
  return 0.5f * v * (1.0f + erff(v * 0.70710678118654752440f));
}

// ---------------------------------------------------------------------------
// Kernel 1: row norms x2[i] = sum_k x[i,k]^2 ; block 0 zeroes md.
// ---------------------------------------------------------------------------
__global__ void k_init(const float* __restrict__ x, float* __restrict__ x2,
                       unsigned* __restrict__ md) {
  const int j = blockIdx.x;
  const int t = threadIdx.x;  // 128 threads
  __shared__ float s[PDIM];
  float v = x[(size_t)j * PDIM + t];
  s[t] = v * v;
  __syncthreads();
  for (int off = 64; off > 0; off >>= 1) {
    if (t < off) s[t] += s[t + off];
    __syncthreads();
  }
  if (t == 0) x2[j] = s[0];
  if (j == 0 && t == 0) *md = 0u;
}

// ---------------------------------------------------------------------------
// Kernel 2: dist = sqrt(max(x2_i + x2_j - 2*(x x^T), 1e-12)) via WMMA f32;
// global max via LDS reduce + atomicMax on float bits (all values >= 0,
// bitwise order-independent -> deterministic).
// Block = 256 threads (8 waves, 4(M) x 2(N)); per-wave tile 32x64;
// block tile 128(M) x 128(N). Grid (32, 32).
// ---------------------------------------------------------------------------
__global__ void k_dist(const float* __restrict__ x, const float* __restrict__ x2,
                       float* __restrict__ dist, unsigned* __restrict__ md) {
  const int lane = threadIdx.x & 31;
  const int wave = threadIdx.x >> 5;
  const int r = lane & 15;
  const int half = lane >> 4;
  const int m_base = blockIdx.y * 128 + (wave >> 1) * 32;
  const int n_base = blockIdx.x * 128 + (wave & 1) * 64;

  v8f c[2][4];
#pragma unroll
  for (int mi = 0; mi < 2; ++mi)
#pragma unroll
    for (int ni = 0; ni < 4; ++ni) c[mi][ni] = v8zero();

  for (int k0 = 0; k0 < PDIM; k0 += 4) {
    v2f a[2], b[4];
#pragma unroll
    for (int mi = 0; mi < 2; ++mi)
      a[mi] = *(const v2f*)(x + (size_t)(m_base + mi * 16 + r) * PDIM + k0 + 2 * half);
#pragma unroll
    for (int ni = 0; ni < 4; ++ni)
      b[ni] = *(const v2f*)(x + (size_t)(n_base + ni * 16 + r) * PDIM + k0 + 2 * half);
#pragma unroll
    for (int mi = 0; mi < 2; ++mi)
#pragma unroll
      for (int ni = 0; ni < 4; ++ni)
        c[mi][ni] = wmma4(a[mi], b[ni], c[mi][ni]);
  }

  float lmax = 0.0f;
#pragma unroll
  for (int mi = 0; mi < 2; ++mi) {
#pragma unroll
    for (int ni = 0; ni < 4; ++ni) {
#pragma unroll
      for (int v = 0; v < 8; ++v) {
        int row = m_base + mi * 16 + v + 8 * half;
        int col = n_base + ni * 16 + r;
        float d2 = x2[row] + x2[col] - 2.0f * c[mi][ni][v];
        float dd = sqrtf(fmaxf(d2, 1e-12f));
        dist[(size_t)row * NPTS + col] = dd;
        lmax = fmaxf(lmax, dd);
      }
    }
  }

  __shared__ float sm[256];
  sm[threadIdx.x] = lmax;
  __syncthreads();
  for (int off = 128; off > 0; off >>= 1) {
    if (threadIdx.x < off) sm[threadIdx.x] = fmaxf(sm[threadIdx.x], sm[threadIdx.x + off]);
    __syncthreads();
  }
  if (threadIdx.x == 0) atomicMax(md, __float_as_uint(sm[0]));
}

// ---------------------------------------------------------------------------
// Kernel 3: alpha = w1(dist, min(r0, md)) elementwise (L2-resident traffic).
// ---------------------------------------------------------------------------
__global__ void k_alpha(const float* __restrict__ dist, float* __restrict__ alpha,
                        const unsigned* __restrict__ mdp, const float* __restrict__ r0p) {
  const float mdv = __uint_as_float(*mdp);
  const float r0e = fminf(*r0p, mdv);
  const size_t total = (size_t)NPTS * NPTS;
  for (size_t idx = (size_t)blockIdx.x * blockDim.x + threadIdx.x; idx < total;
       idx += (size_t)gridDim.x * blockDim.x)
    alpha[idx] = w1f(dist[idx], r0e);
}

// ---------------------------------------------------------------------------
// Column sums: out[j] = sum_i buf[i, j]. Coalesced; prefetch 8 rows ahead
// (speculative global_prefetch_b8 -- OOB tail addresses are dropped).
// ---------------------------------------------------------------------------
__global__ void k_colsum(const float* __restrict__ buf, float* __restrict__ out) {
  const int j = blockIdx.x * blockDim.x + threadIdx.x;
  float s = 0.0f;
  for (int i = 0; i < NPTS; ++i) {
    __builtin_prefetch(buf + (size_t)(i + 8) * NPTS + j, 0, 1);
    s += buf[(size_t)i * NPTS + j];
  }
  out[j] = s;
}

// ---------------------------------------------------------------------------
// Kernel 4/8: split-K partials of A^T @ x (no normalization here).
// A is (n x n) row-major; A^T[m,k] = A[k*n+m].  M=4096, N=128, K=4096/KSPLIT.
// Block = 256 threads (8 waves, 4(M) x 2(N)); per-wave tile 32x64.
// Grid (M/128, KSPLIT). part[kc] gets its own 4096x128 buffer -> no float
// atomics, bit-deterministic.
// ---------------------------------------------------------------------------
__global__ void k_gemm_atx_splitk(const float* __restrict__ A, const float* __restrict__ x,
                                  float* __restrict__ part) {
  const int lane = threadIdx.x & 31;
  const int wave = threadIdx.x >> 5;
  const int r = lane & 15;
  const int half = lane >> 4;
  const int m_base = blockIdx.x * 128 + (wave >> 1) * 32;
  const int n_base = (wave & 1) * 64;
  const int kc = blockIdx.y;
  const int kbeg = kc * (NPTS / KSPLIT);
  const int kend = kbeg + (NPTS / KSPLIT);
  float* __restrict__ outp = part + (size_t)kc * NPTS * PDIM;

  v8f c[2][4];
#pragma unroll
  for (int mi = 0; mi < 2; ++mi)
#pragma unroll
    for (int ni = 0; ni < 4; ++ni) c[mi][ni] = v8zero();

  for (int k0 = kbeg; k0 < kend; k0 += 4) {
    // Prefetch the streamed A operand 64 k-rows ahead (speculative).
    __builtin_prefetch(A + (size_t)(k0 + 64) * NPTS + m_base + r, 0, 1);
    v2f a[2], b[4];
#pragma unroll
    for (int mi = 0; mi < 2; ++mi) {
      const int mrow = m_base + mi * 16 + r;
      a[mi].x = A[(size_t)(k0 + 2 * half) * NPTS + mrow];
      a[mi].y = A[(size_t)(k0 + 2 * half + 1) * NPTS + mrow];
    }
#pragma unroll
    for (int ni = 0; ni < 4; ++ni) {
      const int ncol = n_base + ni * 16 + r;
      b[ni].x = x[(size_t)(k0 + 2 * half) * PDIM + ncol];
      b[ni].y = x[(size_t)(k0 + 2 * half + 1) * PDIM + ncol];
    }
#pragma unroll
    for (int mi = 0; mi < 2; ++mi)
#pragma unroll
      for (int ni = 0; ni < 4; ++ni)
        c[mi][ni] = wmma4(a[mi], b[ni], c[mi][ni]);
  }

#pragma unroll
  for (int mi = 0; mi < 2; ++mi) {
#pragma unroll
    for (int ni = 0; ni < 4; ++ni) {
#pragma unroll
      for (int v = 0; v < 8; ++v) {
        int row = m_base + mi * 16 + v + 8 * half;
        int col = n_base + ni * 16 + r;
        outp[(size_t)row * PDIM + col] = c[mi][ni][v];
      }
    }
  }
}

// ---------------------------------------------------------------------------
// Kernel 5: combine mu partials, normalize, NaN-guard; then
// U = x - mu; u_norm2[j] = |U_j|^2; ydU[j] = <x_j, U_j>.
// ---------------------------------------------------------------------------
__global__ void k_mu_u(const float* __restrict__ part, const float* __restrict__ x,
                       const float* __restrict__ colsum, float* __restrict__ U,
                       float* __restrict__ un2, float* __restrict__ ydU) {
  const int j = blockIdx.x;
  const int t = threadIdx.x;  // 128
  __shared__ float s1[PDIM], s2[PDIM];
  const size_t idx = (size_t)j * PDIM + t;
  float m = 0.0f;
#pragma unroll
  for (int kc = 0; kc < KSPLIT; ++kc) m += part[(size_t)kc * NPTS * PDIM + idx];
  m = m / colsum[j];
  float xv = x[idx];
  if (m != m) m = xv;  // NaN guard -> y
  float u = xv - m;
  U[idx] = u;
  s1[t] = u * u;
  s2[t] = xv * u;
  __syncthreads();
  for (int off = 64; off > 0; off >>= 1) {
    if (t < off) { s1[t] += s1[t + off]; s2[t] += s2[t + off]; }
    __syncthreads();
  }
  if (t == 0) { un2[j] = s1[0]; ydU[j] = s2[0]; }
}

// ---------------------------------------------------------------------------
// Kernel 6: ps = x @ U^T - ydU[j]; fused beta epilogue, written over dist.
// Same tiling as k_dist (M=i, N=j, K=128).
// ---------------------------------------------------------------------------
__global__ void k_ps_beta(const float* __restrict__ x, const float* __restrict__ U,
                          float* __restrict__ db,  // dist in, beta out
                          const float* __restrict__ un2, const float* __restrict__ ydU,
                          const unsigned* __restrict__ mdp,
                          const float* __restrict__ r1p, const float* __restrict__ r2p) {
  const int lane = threadIdx.x & 31;
  const int wave = threadIdx.x >> 5;
  const int r = lane & 15;
  const int half = lane >> 4;
  const int m_base = blockIdx.y * 128 + (wave >> 1) * 32;
  const int n_base = blockIdx.x * 128 + (wave & 1) * 64;

  v8f c[2][4];
#pragma unroll
  for (int mi = 0; mi < 2; ++mi)
#pragma unroll
    for (int ni = 0; ni < 4; ++ni) c[mi][ni] = v8zero();

  for (int k0 = 0; k0 < PDIM; k0 += 4) {
    v2f a[2], b[4];
#pragma unroll
    for (int mi = 0; mi < 2; ++mi)
      a[mi] = *(const v2f*)(x + (size_t)(m_base + mi * 16 + r) * PDIM + k0 + 2 * half);
#pragma unroll
    for (int ni = 0; ni < 4; ++ni)
      b[ni] = *(const v2f*)(U + (size_t)(n_base + ni * 16 + r) * PDIM + k0 + 2 * half);
#pragma unroll
    for (int mi = 0; mi < 2; ++mi)
#pragma unroll
      for (int ni = 0; ni < 4; ++ni)
        c[mi][ni] = wmma4(a[mi], b[ni], c[mi][ni]);
  }

  const float mdv = __uint_as_float(*mdp);
  const float r1e = fminf(*r1p, mdv);
  const float r2e = fminf(*r2p, mdv);

#pragma unroll
  for (int mi = 0; mi < 2; ++mi) {
#pragma unroll
    for (int ni = 0; ni < 4; ++ni) {
#pragma unroll
      for (int v = 0; v < 8; ++v) {
        int row = m_base + mi * 16 + v + 8 * half;  // i
        int col = n_base + ni * 16 + r;             // j
        float ps = c[mi][ni][v] - ydU[col];
        float du2 = fmaxf(ps * ps * un2[col], 1e-6f);
        float d = db[(size_t)row * NPTS + col];
        float dv = sqrtf(fmaxf(d * d - du2, 1e-6f));
        float du = sqrtf(du2);
        db[(size_t)row * NPTS + col] = w2f(dv, r1e) * w2f(du, r2e);
      }
    }
  }
}

// ---------------------------------------------------------------------------
// Kernel 9: combine e_Z partials (normalize + NaN guard), then per-row MLP
// (Linear->GELU->Linear->Linear->GELU->Linear->Sigmoid) + per-row squared error.
// One 128-thread block per row; deterministic (no atomics).
// ---------------------------------------------------------------------------
__global__ void k_mlp(const float* __restrict__ part, const float* __restrict__ colsum,
                      const float* __restrict__ x,
                      const float* __restrict__ We1, const float* __restrict__ be1,
                      const float* __restrict__ We2, const float* __restrict__ be2,
                      const float* __restrict__ Wd1, const float* __restrict__ bd1,
                      const float* __restrict__ Wd2, const float* __restrict__ bd2,
                      float* __restrict__ rowloss) {
  const int j = blockIdx.x;
  const int t = threadIdx.x;  // 128
  __shared__ float y[PDIM], h1[LDIM], z[LDIM], h2[LDIM], red[PDIM];
  const size_t idx = (size_t)j * PDIM + t;
  float yv = 0.0f;
#pragma unroll
  for (int kc = 0; kc < KSPLIT; ++kc) yv += part[(size_t)kc * NPTS * PDIM + idx];
  yv = yv / colsum[j];
  if (yv != yv) yv = x[idx];  // NaN guard -> y
  y[t] = yv;
  __syncthreads();
  if (t < LDIM) {
    float s = be1[t];
    for (int k = 0; k < PDIM; ++k) s += y[k] * We1[t * PDIM + k];
    h1[t] = gelu_exact(s);
  }
  __syncthreads();
  if (t < LDIM) {
    float s = be2[t];
    for (int k = 0; k < LDIM; ++k) s += h1[k] * We2[t * LDIM + k];
    z[t] = s;
  }
  __syncthreads();
  if (t < LDIM) {
    float s = bd1[t];
    for (int k = 0; k < LDIM; ++k) s += z[k] * Wd1[t * LDIM + k];
    h2[t] = gelu_exact(s);
  }
  __syncthreads();
  float s = bd2[t];
  for (int k = 0; k < LDIM; ++k) s += h2[k] * Wd2[t * LDIM + k];
  float xh = 1.0f / (1.0f + expf(-s));
  float d = x[idx] - xh;
  red[t] = d * d;
  __syncthreads();
  for (int off = 64; off > 0; off >>= 1) {
    if (t < off) red[t] += red[t + off];
    __syncthreads();
  }
  if (t == 0) rowloss[j] = red[0];
}

// ---------------------------------------------------------------------------
// Kernel 10: deterministic final reduction of 4096 row losses.
// ---------------------------------------------------------------------------
__global__ void k_final(const float* __restrict__ rowloss, float* __restrict__ out) {
  const int t = threadIdx.x;  // 128
  __shared__ float red[PDIM];
  float s = 0.0f;
  for (int i = t; i < NPTS; i += PDIM) s += rowloss[i];
  red[t] = s;
  __syncthreads();
  for (int off = 64; off > 0; off >>= 1) {
    if (t < off) red[t] += red[t + off];
    __syncthreads();
  }
  if (t == 0) out[0] = red[0] * (1.0f / ((float)NPTS * (float)PDIM));
}

// ---------------------------------------------------------------------------
// Host launcher
// ---------------------------------------------------------------------------
extern "C" void kernel_launch(void* const* d_in, const int* in_sizes, int n_in,
                              void* d_out, int out_size, void* d_ws, size_t ws_size,
                              hipStream_t stream) {
  (void)in_sizes; (void)n_in; (void)out_size; (void)ws_size;

  const float* x   = (const float*)d_in[0];
  const float* r0  = (const float*)d_in[1];
  const float* r1  = (const float*)d_in[2];
  const float* r2  = (const float*)d_in[3];
  const float* We1 = (const float*)d_in[4];
  const float* be1 = (const float*)d_in[5];
  const float* We2 = (const float*)d_in[6];
  const float* be2 = (const float*)d_in[7];
  const float* Wd1 = (const float*)d_in[8];
  const float* bd1 = (const float*)d_in[9];
  const float* Wd2 = (const float*)d_in[10];
  const float* bd2 = (const float*)d_in[11];
  float* out = (float*)d_out;

  float* ws = (float*)d_ws;
  size_t off = 0;
  float* dist  = ws + off; off += (size_t)NPTS * NPTS;  // later reused as beta
  float* alpha = ws + off; off += (size_t)NPTS * NPTS;
  float* part  = ws + off; off += (size_t)KSPLIT * NPTS * PDIM;  // split-K partials
  float* x2    = ws + off; off += NPTS;
  float* colA  = ws + off; off += NPTS;
  float* colB  = ws + off; off += NPTS;
  float* Um    = ws + off; off += (size_t)NPTS * PDIM;
  float* un2   = ws + off; off += NPTS;
  float* ydU   = ws + off; off += NPTS;
  float* rowl  = ws + off; off += NPTS;
  unsigned* md = (unsigned*)(ws + off); off += 1;

  k_init<<<NPTS, 128, 0, stream>>>(x, x2, md);
  k_dist<<<dim3(NPTS / 128, NPTS / 128), 256, 0, stream>>>(x, x2, dist, md);
  k_alpha<<<8192, 256, 0, stream>>>(dist, alpha, md, r0);
  k_colsum<<<NPTS / 128, 128, 0, stream>>>(alpha, colA);
  k_gemm_atx_splitk<<<dim3(NPTS / 128, KSPLIT), 256, 0, stream>>>(alpha, x, part);
  k_mu_u<<<NPTS, 128, 0, stream>>>(part, x, colA, Um, un2, ydU);
  k_ps_beta<<<dim3(NPTS / 128, NPTS / 128), 256, 0, stream>>>(x, Um, dist, un2, ydU, md, r1, r2);
  k_colsum<<<NPTS / 128, 128, 0, stream>>>(dist, colB);  // dist now holds beta
  k_gemm_atx_splitk<<<dim3(NPTS / 128, KSPLIT), 256, 0, stream>>>(dist, x, part);
  k_mlp<<<NPTS, 128, 0, stream>>>(part, colB, x, We1, be1, We2, be2, Wd1, bd1, Wd2, bd2, rowl);
  k_final<<<1, 128, 0, stream>>>(rowl, out);
}